// VisionExpertMLP_49855980372282
// MI455X (gfx1250) — compile-verified
//
#include <hip/hip_runtime.h>

// ---------------------------------------------------------------------------
// VisionExpertMLP for MI455X (gfx1250), fp32 WMMA path.
//
// Weight-bandwidth bound: 1.08 GB fp32 weights @ 23.3 TB/s ≈ 52 us floor;
// fp32 WMMA compute ~4 us. Strategy:
//   * compute BOTH experts per token tile, select per row (weights still read
//     once grid-wide; avoids MoE gather/scatter).
//   * token tile = blockIdx.x (fast dispatch dim) so all token-blocks of one
//     F-slice run concurrently and share the weight stream through L2 ->
//     each weight element fetched from HBM ~once.
//   * 64-token tile per block: each weight fragment feeds 4 WMMAs
//     (4 row-subtiles; 16 per K-step over 4 matrices) -> wmma-dominated inner loop.
// ---------------------------------------------------------------------------

typedef __attribute__((ext_vector_type(2))) float v2f;
typedef __attribute__((ext_vector_type(8))) float v8f;

#define D_HID 4096
#define D_INT 11008
#define N_TOK 4096          // B*L = 2*2048
#define KC    64            // K chunk staged in LDS
#define XS    68            // LDS row stride (floats): 16B-aligned, conflict-free
#define TT    64            // tokens per block (4 WMMA row-subtiles)

#define WMMA_F32(A, B, C) \
  __builtin_amdgcn_wmma_f32_16x16x4_f32(false, (A), false, (B), (short)0, (C), false, false)

// --------------------------- mask kernel -----------------------------------
__global__ void __launch_bounds__(256)
vmlp_mask(const int* __restrict__ tt, int* __restrict__ mask, int Bn, int Ln) {
  int idx = blockIdx.x * blockDim.x + threadIdx.x;
  if (idx >= Bn * Ln) return;
  int i = idx % Ln;
  mask[idx] = (i + 1 < Ln) && (tt[idx] == 1) && (tt[idx + 1] == 1);
}

// --------------------------- gate/up kernel --------------------------------
// Block = 256 threads = 8 waves; tile = 64 tokens x 128 F-cols.
__global__ void __launch_bounds__(256)
vmlp_gateup(const float* __restrict__ X,
            const float* __restrict__ Lg, const float* __restrict__ Lu,
            const float* __restrict__ Vg, const float* __restrict__ Vu,
            const int* __restrict__ mask, float* __restrict__ H) {
  __shared__ float xs[TT * XS];
  __shared__ int   ms[TT];

  const int t0   = blockIdx.x * TT;                               // token tile (fast)
  const int n0   = (blockIdx.y << 7) + ((threadIdx.x >> 5) << 4); // wave's F cols
  const int lane = threadIdx.x & 31;
  const int m    = lane & 15;           // A row (mod 16) / B col
  const int kk   = (lane >> 4) << 1;    // K sub-offset: 0 / 2
  const int lrow = threadIdx.x >> 2;          // loader row 0..63
  const int lseg = (threadIdx.x & 3) << 4;    // loader col 0,16,32,48

  if (threadIdx.x < TT) ms[threadIdx.x] = mask[t0 + threadIdx.x];

  v8f a_lg[4] = {}; v8f a_lu[4] = {}; v8f a_vg[4] = {}; v8f a_vu[4] = {};

  for (int k0 = 0; k0 < D_HID; k0 += KC) {
    __syncthreads();
    // stage X[t0..t0+63][k0..k0+63] -> LDS (4x float4 per thread)
    const float* src = X + (size_t)(t0 + lrow) * D_HID + k0 + lseg;
    float*       dst = xs + lrow * XS + lseg;
    *(float4*)(dst)      = *(const float4*)(src);
    *(float4*)(dst + 4)  = *(const float4*)(src + 4);
    *(float4*)(dst + 8)  = *(const float4*)(src + 8);
    *(float4*)(dst + 12) = *(const float4*)(src + 12);
    if (k0 + KC < D_HID) __builtin_prefetch(src + KC, 0, 0);
    __syncthreads();

#pragma unroll 2
    for (int s = 0; s < KC; s += 4) {
      // A fragments for the 4 row-subtiles
      const v2f a0 = *(const v2f*)(xs + (m)      * XS + s + kk);
      const v2f a1 = *(const v2f*)(xs + (16 + m) * XS + s + kk);
      const v2f a2 = *(const v2f*)(xs + (32 + m) * XS + s + kk);
      const v2f a3 = *(const v2f*)(xs + (48 + m) * XS + s + kk);
      // B fragment (4x16): lane col n=m holds K rows {kk, kk+1}
      const size_t wo = (size_t)(k0 + s + kk) * D_INT + n0 + m;
      v2f b;
      b.x = Lg[wo]; b.y = Lg[wo + D_INT];
      a_lg[0] = WMMA_F32(a0, b, a_lg[0]);
      a_lg[1] = WMMA_F32(a1, b, a_lg[1]);
      a_lg[2] = WMMA_F32(a2, b, a_lg[2]);
      a_lg[3] = WMMA_F32(a3, b, a_lg[3]);
      b.x = Lu[wo]; b.y = Lu[wo + D_INT];
      a_lu[0] = WMMA_F32(a0, b, a_lu[0]);
      a_lu[1] = WMMA_F32(a1, b, a_lu[1]);
      a_lu[2] = WMMA_F32(a2, b, a_lu[2]);
      a_lu[3] = WMMA_F32(a3, b, a_lu[3]);
      b.x = Vg[wo]; b.y = Vg[wo + D_INT];
      a_vg[0] = WMMA_F32(a0, b, a_vg[0]);
      a_vg[1] = WMMA_F32(a1, b, a_vg[1]);
      a_vg[2] = WMMA_F32(a2, b, a_vg[2]);
      a_vg[3] = WMMA_F32(a3, b, a_vg[3]);
      b.x = Vu[wo]; b.y = Vu[wo + D_INT];
      a_vu[0] = WMMA_F32(a0, b, a_vu[0]);
      a_vu[1] = WMMA_F32(a1, b, a_vu[1]);
      a_vu[2] = WMMA_F32(a2, b, a_vu[2]);
      a_vu[3] = WMMA_F32(a3, b, a_vu[3]);
    }
  }

  // C/D layout: VGPR r -> row r + 8*(lane>=16), col n = lane&15
  const int half = (lane >> 4) << 3;
#pragma unroll
  for (int st = 0; st < 4; ++st) {
#pragma unroll
    for (int r = 0; r < 8; ++r) {
      const int trow = (st << 4) + r + half;
      const bool vis = ms[trow] != 0;
      const float g = vis ? a_vg[st][r] : a_lg[st][r];
      const float u = vis ? a_vu[st][r] : a_lu[st][r];
      const float h = (g / (1.0f + __expf(-g))) * u;   // silu(g) * u
      H[(size_t)(t0 + trow) * D_INT + n0 + m] = h;
    }
  }
}

// --------------------------- down kernel -----------------------------------
// out[t, d] = H[t, :] @ Wd[expert(t)]; K = 11008 = 172 * 64.
__global__ void __launch_bounds__(256)
vmlp_down(const float* __restrict__ H,
          const float* __restrict__ Ld, const float* __restrict__ Vd,
          const int* __restrict__ mask, float* __restrict__ out) {
  __shared__ float xs[TT * XS];
  __shared__ int   ms[TT];

  const int t0   = blockIdx.x * TT;
  const int n0   = (blockIdx.y << 7) + ((threadIdx.x >> 5) << 4);
  const int lane = threadIdx.x & 31;
  const int m    = lane & 15;
  const int kk   = (lane >> 4) << 1;
  const int lrow = threadIdx.x >> 2;
  const int lseg = (threadIdx.x & 3) << 4;

  if (threadIdx.x < TT) ms[threadIdx.x] = mask[t0 + threadIdx.x];

  v8f a_l[4] = {}; v8f a_v[4] = {};

  for (int k0 = 0; k0 < D_INT; k0 += KC) {
    __syncthreads();
    const float* src = H + (size_t)(t0 + lrow) * D_INT + k0 + lseg;
    float*       dst = xs + lrow * XS + lseg;
    *(float4*)(dst)      = *(const float4*)(src);
    *(float4*)(dst + 4)  = *(const float4*)(src + 4);
    *(float4*)(dst + 8)  = *(const float4*)(src + 8);
    *(float4*)(dst + 12) = *(const float4*)(src + 12);
    if (k0 + KC < D_INT) __builtin_prefetch(src + KC, 0, 0);
    __syncthreads();

#pragma unroll 2
    for (int s = 0; s < KC; s += 4) {
      const v2f a0 = *(const v2f*)(xs + (m)      * XS + s + kk);
      const v2f a1 = *(const v2f*)(xs + (16 + m) * XS + s + kk);
      const v2f a2 = *(const v2f*)(xs + (32 + m) * XS + s + kk);
      const v2f a3 = *(const v2f*)(xs + (48 + m) * XS + s + kk);
      const size_t wo = (size_t)(k0 + s + kk) * D_HID + n0 + m;
      v2f b;
      b.x = Ld[wo]; b.y = Ld[wo + D_HID];
      a_l[0] = WMMA_F32(a0, b, a_l[0]);
      a_l[1] = WMMA_F32(a1, b, a_l[1]);
      a_l[2] = WMMA_F32(a2, b, a_l[2]);
      a_l[3] = WMMA_F32(a3, b, a_l[3]);
      b.x = Vd[wo]; b.y = Vd[wo + D_HID];
      a_v[0] = WMMA_F32(a0, b, a_v[0]);
      a_v[1] = WMMA_F32(a1, b, a_v[1]);
      a_v[2] = WMMA_F32(a2, b, a_v[2]);
      a_v[3] = WMMA_F32(a3, b, a_v[3]);
    }
  }

  const int half = (lane >> 4) << 3;
#pragma unroll
  for (int st = 0; st < 4; ++st) {
#pragma unroll
    for (int r = 0; r < 8; ++r) {
      const int trow = (st << 4) + r + half;
      out[(size_t)(t0 + trow) * D_HID + n0 + m] =
          (ms[trow] != 0) ? a_v[st][r] : a_l[st][r];
    }
  }
}

// --------------------------- launch ----------------------------------------
extern "C" void kernel_launch(void* const* d_in, const int* in_sizes, int n_in,
                              void* d_out, int out_size, void* d_ws, size_t ws_size,
                              hipStream_t stream) {
  (void)in_sizes; (void)n_in; (void)out_size; (void)ws_size;

  const float* X  = (const float*)d_in[0];  // hidden_states (2,2048,4096)
  const int*   tt = (const int*)d_in[1];    // token_type_ids (2,2048)
  const float* Lg = (const float*)d_in[2];  // lang_gate (4096,11008)
  const float* Lu = (const float*)d_in[3];  // lang_up   (4096,11008)
  const float* Ld = (const float*)d_in[4];  // lang_down (11008,4096)
  const float* Vg = (const float*)d_in[5];  // vis_gate
  const float* Vu = (const float*)d_in[6];  // vis_up
  const float* Vd = (const float*)d_in[7];  // vis_down
  float* out = (float*)d_out;

  // workspace: [mask: N_TOK ints][pad to 256B][H: N_TOK x D_INT fp32 ≈ 180.5 MB]
  int*   mask = (int*)d_ws;
  float* H    = (float*)((char*)d_ws + (((size_t)N_TOK * sizeof(int) + 255) & ~(size_t)255));

  vmlp_mask<<<(N_TOK + 255) / 256, 256, 0, stream>>>(tt, mask, 2, 2048);

  // token tiles on x (fast dispatch dim) -> concurrent blocks share one
  // F-slice weight stream through L2.
  dim3 gA(N_TOK / TT, D_INT / 128);   // 64 x 86
  vmlp_gateup<<<gA, 256, 0, stream>>>(X, Lg, Lu, Vg, Vu, mask, H);

  dim3 gB(N_TOK / TT, D_HID / 128);   // 64 x 32
  vmlp_down<<<gB, 256, 0, stream>>>(H, Ld, Vd, mask, out);
}